// AttentionModel_8495445311628
// MI455X (gfx1250) — compile-verified
//
#include <hip/hip_runtime.h>
#include <math.h>

// ---------------------------------------------------------------------------
// AttentionModel greedy decoder for MI455X (gfx1250, wave32, WMMA).
//   B=128, N=512, D=128, H=8, DH=16
// Phase 1: proj = emb @ W_node_proj via V_WMMA_F32_16X16X4_F32 (A in regs,
//          read exactly once; W is L2-resident).
// Phase 2: fixed_ctx = mean_n(emb) @ W_fixed_ctx.
// Phase 3: one 512-thread WG (16 wave32's) per batch element runs all 512
//          greedy decode steps locally. Step tail uses one fused
//          argmax+sum-exp reduction (tanh clip bounds logits in [-10,10],
//          so sum(exp(logit)) is fp32-safe without max subtraction).
// ---------------------------------------------------------------------------

#define BB   128
#define NN   512
#define DD   128
#define HH   8
#define DHH  16
#define D3   384
#define FCLIP 10.0f
#define FNEG  (-1.0e9f)

typedef __attribute__((ext_vector_type(2))) float v2f;
typedef __attribute__((ext_vector_type(8))) float v8f;

// ---------------------------------------------------------------------------
// Kernel 1: proj[BN,384] = A[BN,128] * W[128,384], fp32 WMMA 16x16x4.
// 8 waves / block, one 16-row block each; wave loops over the 24 column tiles.
// ---------------------------------------------------------------------------
__global__ __launch_bounds__(256) void proj_gemm_wmma(
    const float* __restrict__ A, const float* __restrict__ W,
    float* __restrict__ C)
{
    const int lane = threadIdx.x & 31;
    const int wave = threadIdx.x >> 5;
    const int mt = blockIdx.x * 8 + wave;     // 512 blocks * 8 waves = 4096 tiles
    const int m0 = mt * 16;
    const int lm = lane & 15;
    const int koff = (lane >> 4) * 2;

    // Entire 16x128 A row-block in registers (64 VGPRs).
    v2f afr[32];
    const float* Arow = A + (size_t)(m0 + lm) * DD;
#pragma unroll
    for (int ks = 0; ks < 32; ++ks) {
        float2 av = *(const float2*)(Arow + ks * 4 + koff);
        afr[ks].x = av.x; afr[ks].y = av.y;
    }

    const int rbase = m0 + ((lane >> 4) << 3);
    for (int nt = 0; nt < 24; ++nt) {
        const int n0 = nt * 16;
        v8f acc = {};
#pragma unroll
        for (int ks = 0; ks < 32; ++ks) {
            const int k = ks * 4 + koff;
            v2f bf;
            bf.x = W[(size_t)k * D3 + n0 + lm];
            bf.y = W[(size_t)(k + 1) * D3 + n0 + lm];
            acc = __builtin_amdgcn_wmma_f32_16x16x4_f32(
                false, afr[ks], false, bf, (short)0, acc, false, false);
        }
#pragma unroll
        for (int v = 0; v < 8; ++v)
            C[(size_t)(rbase + v) * D3 + n0 + lm] = acc[v];
    }
}

// ---------------------------------------------------------------------------
// Kernel 2: fixed_ctx[b,:] = mean_n(emb[b,n,:]) @ W_fixed_ctx. 1 WG / b.
// ---------------------------------------------------------------------------
__global__ __launch_bounds__(128) void fixed_ctx_kernel(
    const float* __restrict__ emb, const float* __restrict__ Wf,
    float* __restrict__ fixedc)
{
    __shared__ float ge[DD];
    const int b = blockIdx.x, d = threadIdx.x;
    const float* p = emb + (size_t)b * NN * DD + d;
    float s = 0.f;
    for (int n = 0; n < NN; ++n) s += p[(size_t)n * DD];
    ge[d] = s * (1.0f / NN);
    __syncthreads();
    float acc = 0.f;
    for (int k = 0; k < DD; ++k) acc += ge[k] * Wf[(size_t)k * DD + d];
    fixedc[(size_t)b * DD + d] = acc;
}

// ---------------------------------------------------------------------------
// Kernel 3: full 512-step greedy decode, one 512-thread WG per batch element.
// 16 wave32's; waves (2h, 2h+1) own head h (n-range split in halves).
// ---------------------------------------------------------------------------
__global__ __launch_bounds__(512) void decode_kernel(
    const float* __restrict__ emb,    // [B,N,D]
    const float* __restrict__ proj,   // [B,N,3D] : gK | gV | logit_K
    const float* __restrict__ fixedc, // [B,D]
    const float* __restrict__ Wstep,  // [2D,D]
    const float* __restrict__ Wout,   // [D,D]
    const float* __restrict__ Wph,    // [2D]
    float* __restrict__ logp_out,     // [B,N,N]
    float* __restrict__ pi_out,       // [B,N]
    float* __restrict__ ll_out)       // [B]
{
    __shared__ float sm_attn[HH * NN];   // compat -> attn (16 KB)
    __shared__ float sm_part[16 * DHH];  // per-wave reduced head partials
    __shared__ float sm_logits[NN];
    __shared__ float sm_q[DD];
    __shared__ float sm_glimpse[DD];
    __shared__ float sm_ctx[2 * DD];
    __shared__ float sm_qp[4 * DD];      // split-K partials for GEMVs
    __shared__ int   sm_visited[NN];
    __shared__ float sm_wredA[16];       // per-wave max (softmax) / max (final)
    __shared__ float sm_wredB[16];       // per-wave sum (softmax) / sumexp (final)
    __shared__ int   sm_wredI[16];       // per-wave argmax index (final)
    __shared__ int   sm_state[2];        // first, prev

    const int b = blockIdx.x;
    const int tid = threadIdx.x;
    const int lane = tid & 31;
    const int wave = tid >> 5;
    const float inv_sqrt_d = 0.08838834764831845f; // 1/sqrt(128)

    for (int n = tid; n < NN; n += 512) sm_visited[n] = 0;
    if (tid == 0) { sm_state[0] = 0; sm_state[1] = 0; }
    __syncthreads();

    const float* projb = proj + (size_t)b * NN * D3;
    const float* embb  = emb  + (size_t)b * NN * DD;
    float ll_acc = 0.f;                  // identical in every thread

    for (int t = 0; t < NN; ++t) {
        // ---- step context ----
        if (tid < 2 * DD) {
            float c;
            if (t == 0) c = Wph[tid];
            else {
                int node = (tid < DD) ? sm_state[0] : sm_state[1];
                c = embb[(size_t)node * DD + (tid & (DD - 1))];
            }
            sm_ctx[tid] = c;
        }
        __syncthreads();

        // ---- q = fixed_ctx + step_ctx @ W_step  (split-K over 4 parts) ----
        {
            const int part = tid >> 7, d = tid & (DD - 1);
            float acc = 0.f;
            const int k0 = part * 64;
            for (int k = k0; k < k0 + 64; ++k)
                acc += sm_ctx[k] * Wstep[(size_t)k * DD + d];
            sm_qp[part * DD + d] = acc;
        }
        __syncthreads();
        if (tid < DD)
            sm_q[tid] = fixedc[(size_t)b * DD + tid] + sm_qp[tid] +
                        sm_qp[DD + tid] + sm_qp[2 * DD + tid] + sm_qp[3 * DD + tid];
        __syncthreads();

        // ---- compat[h,n] = <qh, gK[n,h,:]> / sqrt(DH), masked ----
        for (int idx = tid; idx < NN * HH; idx += 512) {
            const int n = idx >> 3, h = idx & 7;
            const float4* g  = (const float4*)(projb + (size_t)n * D3 + h * DHH);
            const float4* qv = (const float4*)(sm_q + h * DHH);
            float dot = 0.f;
#pragma unroll
            for (int j = 0; j < 4; ++j) {
                float4 gv = g[j], qq = qv[j];
                dot += gv.x*qq.x + gv.y*qq.y + gv.z*qq.z + gv.w*qq.w;
            }
            sm_attn[h * NN + n] = sm_visited[n] ? FNEG : dot * 0.25f;
        }
        __syncthreads();

        // ---- per-head softmax + weighted-V (waves 2h,2h+1 share head h) ----
        {
            const int h = wave >> 1;
            const int nbase = (wave & 1) * 256;
            float mx = FNEG;
#pragma unroll
            for (int i = 0; i < 8; ++i)
                mx = fmaxf(mx, sm_attn[h * NN + nbase + lane + i * 32]);
            for (int off = 16; off > 0; off >>= 1)
                mx = fmaxf(mx, __shfl_xor(mx, off, 32));
            if (lane == 0) sm_wredA[wave] = mx;
            __syncthreads();
            mx = fmaxf(sm_wredA[2 * h], sm_wredA[2 * h + 1]);

            float sum = 0.f;
#pragma unroll
            for (int i = 0; i < 8; ++i) {
                const int n = nbase + lane + i * 32;
                float e = __expf(sm_attn[h * NN + n] - mx);
                sm_attn[h * NN + n] = e;
                sum += e;
            }
            for (int off = 16; off > 0; off >>= 1)
                sum += __shfl_xor(sum, off, 32);
            if (lane == 0) sm_wredB[wave] = sum;
            __syncthreads();
            const float inv = 1.0f / (sm_wredB[2 * h] + sm_wredB[2 * h + 1]);

            float acc[DHH];
#pragma unroll
            for (int j = 0; j < DHH; ++j) acc[j] = 0.f;
#pragma unroll
            for (int i = 0; i < 8; ++i) {
                const int n = nbase + lane + i * 32;
                const float a = sm_attn[h * NN + n] * inv;
                const float4* g = (const float4*)(projb + (size_t)n * D3 + DD + h * DHH);
#pragma unroll
                for (int j = 0; j < 4; ++j) {
                    float4 gv = g[j];
                    acc[4*j+0] += a * gv.x; acc[4*j+1] += a * gv.y;
                    acc[4*j+2] += a * gv.z; acc[4*j+3] += a * gv.w;
                }
            }
            // in-wave butterfly: all lanes end with the full 16 sums
#pragma unroll
            for (int off = 16; off > 0; off >>= 1) {
#pragma unroll
                for (int j = 0; j < DHH; ++j)
                    acc[j] += __shfl_xor(acc[j], off, 32);
            }
            if (lane == 0) {
#pragma unroll
                for (int j = 0; j < DHH; ++j) sm_part[wave * DHH + j] = acc[j];
            }
        }
        __syncthreads();

        // ---- glimpse = heads @ W_out (split-K; heads folded from sm_part) ----
        {
            const int part = tid >> 7, d = tid & (DD - 1);
            float acc = 0.f;
            const int k0 = part * 32;
#pragma unroll
            for (int k = k0; k < k0 + 32; ++k) {
                const int h2 = (k >> 4) * 2, kk = k & 15;
                const float hk = sm_part[h2 * DHH + kk] + sm_part[(h2 + 1) * DHH + kk];
                acc += hk * Wout[(size_t)k * DD + d];
            }
            sm_qp[part * DD + d] = acc;
        }
        __syncthreads();
        if (tid < DD)
            sm_glimpse[tid] = sm_qp[tid] + sm_qp[DD + tid] +
                              sm_qp[2 * DD + tid] + sm_qp[3 * DD + tid];
        __syncthreads();

        // ---- logits[n] = CLIP * tanh(<glimpse, logit_K[n]> / sqrt(D)) ----
        float v;  // this thread's logit (n == tid), kept in a register
        {
            const float4* g  = (const float4*)(projb + (size_t)tid * D3 + 2 * DD);
            const float4* gl = (const float4*)sm_glimpse;
            float dot = 0.f;
#pragma unroll
            for (int j = 0; j < DD / 4; ++j) {
                float4 a = g[j], c = gl[j];
                dot += a.x*c.x + a.y*c.y + a.z*c.z + a.w*c.w;
            }
            float L = FCLIP * tanhf(dot * inv_sqrt_d);
            v = sm_visited[tid] ? FNEG : L;
            sm_logits[tid] = v;          // visibility covered by next barrier
        }

        // ---- fused argmax + sum-of-exp (logits in [-10,10]; exp(-1e9)=0) ----
        {
            float bv = v; int bi = tid;
            float es = __expf(v);
            for (int off = 16; off > 0; off >>= 1) {
                const float ov = __shfl_xor(bv, off, 32);
                const int   oi = __shfl_xor(bi, off, 32);
                es += __shfl_xor(es, off, 32);
                if (ov > bv || (ov == bv && oi < bi)) { bv = ov; bi = oi; }
            }
            if (lane == 0) {
                sm_wredA[wave] = bv; sm_wredI[wave] = bi; sm_wredB[wave] = es;
            }
        }
        __syncthreads();
        float gv = sm_wredA[0]; int sel = sm_wredI[0]; float gs = sm_wredB[0];
#pragma unroll
        for (int w = 1; w < 16; ++w) {
            const float ov = sm_wredA[w]; const int oi = sm_wredI[w];
            gs += sm_wredB[w];
            if (ov > gv || (ov == gv && oi < sel)) { gv = ov; sel = oi; }
        }
        const float lse = __logf(gs);

        // ---- write log_p row, update state ----
        logp_out[((size_t)b * NN + t) * NN + tid] = v - lse;
        ll_acc += sm_logits[sel] - lse;      // broadcast LDS read, same in all threads
        if (tid == 0) {
            pi_out[(size_t)b * NN + t] = (float)sel;
            sm_visited[sel] = 1;
            if (t == 0) sm_state[0] = sel;
            sm_state[1] = sel;
        }
        __syncthreads();
    }
    if (tid == 0) ll_out[b] = ll_acc;
}

// ---------------------------------------------------------------------------
extern "C" void kernel_launch(void* const* d_in, const int* in_sizes, int n_in,
                              void* d_out, int out_size, void* d_ws, size_t ws_size,
                              hipStream_t stream) {
    const float* emb   = (const float*)d_in[0]; // [B,N,D]
    const float* Wnode = (const float*)d_in[1]; // [D,3D]
    const float* Wfix  = (const float*)d_in[2]; // [D,D]
    const float* Wstep = (const float*)d_in[3]; // [2D,D]
    const float* Wout  = (const float*)d_in[4]; // [D,D]
    const float* Wph   = (const float*)d_in[5]; // [2D]

    float* out  = (float*)d_out;
    float* logp = out;                               // B*N*N
    float* pi   = logp + (size_t)BB * NN * NN;       // B*N
    float* ll   = pi + (size_t)BB * NN;              // B

    float* proj   = (float*)d_ws;                    // B*N*3D floats (~96 MB)
    float* fixedc = proj + (size_t)BB * NN * D3;     // B*D floats

    proj_gemm_wmma<<<dim3(BB * NN / (16 * 8)), dim3(256), 0, stream>>>(
        emb, Wnode, proj);
    fixed_ctx_kernel<<<dim3(BB), dim3(128), 0, stream>>>(emb, Wfix, fixedc);
    decode_kernel<<<dim3(BB), dim3(512), 0, stream>>>(
        emb, proj, fixedc, Wstep, Wout, Wph, logp, pi, ll);
}